// Layer_42657615184403
// MI455X (gfx1250) — compile-verified
//
#include <hip/hip_runtime.h>
#include <math.h>

#ifndef M_PI
#define M_PI 3.14159265358979323846
#endif

#define RES_B 20
#define RES_A 19
#define NGRID (RES_B * RES_A)

typedef float v2f __attribute__((ext_vector_type(2)));
typedef float v8f __attribute__((ext_vector_type(8)));

#if __has_builtin(__builtin_amdgcn_wmma_f32_16x16x4_f32)
#define USE_WMMA_F32 1
#endif

// ---------------------------------------------------------------------------
// Kernel 1: build Gaunt coefficients G[9][9][9] on device (deterministic).
// Gauss-Legendre (n=20) via Newton in double, normalized assoc. Legendre up
// to l=2, real SH on the 20x19 grid, quadrature-weighted triple product.
// ---------------------------------------------------------------------------
__global__ void init_basis_kernel(float* __restrict__ G) {
    __shared__ double Ysh[9][NGRID];
    __shared__ double QWsh[NGRID];
    __shared__ double xg[RES_B], wg[RES_B];
    const int t = threadIdx.x;

    if (t < RES_B) {
        const int n = RES_B;
        double x = cos(M_PI * (t + 0.75) / (n + 0.5));
        for (int it = 0; it < 100; ++it) {
            double p0 = 1.0, p1 = x;
            for (int k = 2; k <= n; ++k) {
                double p2 = ((2.0 * k - 1.0) * x * p1 - (k - 1.0) * p0) / (double)k;
                p0 = p1; p1 = p2;
            }
            double dp = n * (x * p1 - p0) / (x * x - 1.0);
            double dx = p1 / dp;
            x -= dx;
            if (fabs(dx) < 1e-15) break;
        }
        double p0 = 1.0, p1 = x;
        for (int k = 2; k <= n; ++k) {
            double p2 = ((2.0 * k - 1.0) * x * p1 - (k - 1.0) * p0) / (double)k;
            p0 = p1; p1 = p2;
        }
        double dp = n * (x * p1 - p0) / (x * x - 1.0);
        xg[t] = x;
        wg[t] = 2.0 / ((1.0 - x * x) * dp * dp);
    }
    __syncthreads();

    for (int idx = t; idx < NGRID; idx += blockDim.x) {
        int b = idx / RES_A, a = idx % RES_A;
        double x = xg[b];
        double s = sqrt(fmax(0.0, 1.0 - x * x));
        double alpha = 2.0 * M_PI * (double)a / (double)RES_A;
        double P00 = sqrt(1.0 / (4.0 * M_PI));
        double P11 = -sqrt(3.0 / 2.0) * s * P00;
        double P22 = -sqrt(5.0 / 4.0) * s * P11;
        double P10 = sqrt(3.0) * x * P00;
        double P21 = sqrt(5.0) * x * P11;
        double P20 = (sqrt(15.0) * 0.5) * (x * P10 - sqrt(1.0 / 3.0) * P00);
        double r2 = sqrt(2.0);
        Ysh[0][idx] = P00;
        Ysh[1][idx] = r2 * P11 * sin(alpha);        // l=1, m=-1
        Ysh[2][idx] = P10;                          // l=1, m=0
        Ysh[3][idx] = r2 * P11 * cos(alpha);        // l=1, m=+1
        Ysh[4][idx] = r2 * P22 * sin(2.0 * alpha);  // l=2, m=-2
        Ysh[5][idx] = r2 * P21 * sin(alpha);        // l=2, m=-1
        Ysh[6][idx] = P20;                          // l=2, m=0
        Ysh[7][idx] = r2 * P21 * cos(alpha);        // l=2, m=+1
        Ysh[8][idx] = r2 * P22 * cos(2.0 * alpha);  // l=2, m=+2
        QWsh[idx] = wg[b] * (2.0 * M_PI / (double)RES_A);
    }
    __syncthreads();

    for (int kk = t; kk < 729; kk += blockDim.x) {
        int k1 = kk / 81, k2 = (kk / 9) % 9, k3 = kk % 9;
        double acc = 0.0;
        for (int idx = 0; idx < NGRID; ++idx)
            acc += Ysh[k1][idx] * Ysh[k2][idx] * Ysh[k3][idx] * QWsh[idx];
        G[kk] = (float)acc;
    }
}

// ---------------------------------------------------------------------------
// Kernel 2: per-edge message construction + scatter-add (L2-resident atomics).
// agg layout per node (267 floats): [0:67]=l0, [67:172]=l1 (m*3+i), [172:267]=l2 (m*5+i)
// ---------------------------------------------------------------------------
__global__ void edge_kernel(const float* __restrict__ x, const float* __restrict__ pos,
                            const float* __restrict__ Wx, const float* __restrict__ Wy,
                            const int* __restrict__ senders, const int* __restrict__ receivers,
                            const float* __restrict__ G, float* __restrict__ agg, int E) {
    __shared__ float Gs[729];
    __shared__ float Wxs[336];
    __shared__ float Wys[6];
    for (int i = threadIdx.x; i < 729; i += blockDim.x) Gs[i] = G[i];
    for (int i = threadIdx.x; i < 336; i += blockDim.x) Wxs[i] = Wx[i];
    if (threadIdx.x < 6) Wys[threadIdx.x] = Wy[threadIdx.x];
    __syncthreads();

    int e = blockIdx.x * blockDim.x + threadIdx.x;
    if (e >= E) return;
    int s = senders[e], r = receivers[e];
    const float* xs = x + (size_t)s * 240;
    float* ag = agg + (size_t)r * 267;
    const float inv12 = 1.0f / 12.0f;

    // unit edge vector + real SH l=1,2
    float rx = pos[3 * r + 0] - pos[3 * s + 0];
    float ry = pos[3 * r + 1] - pos[3 * s + 1];
    float rz = pos[3 * r + 2] - pos[3 * s + 2];
    float rn = sqrtf(rx * rx + ry * ry + rz * rz) + 1e-12f;
    float X = rx / rn, Y = ry / rn, Z = rz / rn;
    const float c1 = 0.48860251190291992f;   // sqrt(3/4pi)
    const float c2 = 1.09254843059207907f;   // sqrt(15/4pi)
    const float c20 = 0.31539156525252005f;  // sqrt(5/16pi)
    float sh[9];
    sh[0] = 0.0f;
    sh[1] = c1 * Y; sh[2] = c1 * Z; sh[3] = c1 * X;
    sh[4] = c2 * X * Y; sh[5] = c2 * Y * Z; sh[6] = c20 * (3.0f * Z * Z - 1.0f);
    sh[7] = c2 * X * Z; sh[8] = 0.5f * c2 * (X * X - Y * Y);

    float cx[3][9];
#pragma unroll
    for (int ch = 0; ch < 3; ++ch)
#pragma unroll
        for (int k = 0; k < 9; ++k) cx[ch][k] = 0.0f;

    // l=0 block: 64 channels
    for (int c = 0; c < 64; ++c) {
        float v = xs[c];
        atomicAdd(&ag[c], v * inv12);
        cx[0][0] += v * Wxs[c];
        cx[1][0] += v * Wxs[112 + c];
        cx[2][0] += v * Wxs[224 + c];
    }
    // l=1 block: 32 x 3
    for (int m = 0; m < 32; ++m) {
        float w0 = Wxs[64 + m], w1 = Wxs[112 + 64 + m], w2 = Wxs[224 + 64 + m];
#pragma unroll
        for (int i = 0; i < 3; ++i) {
            float v = xs[64 + 3 * m + i];
            atomicAdd(&ag[67 + 3 * m + i], v * inv12);
            cx[0][1 + i] += v * w0; cx[1][1 + i] += v * w1; cx[2][1 + i] += v * w2;
        }
    }
    // l=2 block: 16 x 5
    for (int m = 0; m < 16; ++m) {
        float w0 = Wxs[96 + m], w1 = Wxs[112 + 96 + m], w2 = Wxs[224 + 96 + m];
#pragma unroll
        for (int i = 0; i < 5; ++i) {
            float v = xs[160 + 5 * m + i];
            atomicAdd(&ag[172 + 5 * m + i], v * inv12);
            cx[0][4 + i] += v * w0; cx[1][4 + i] += v * w1; cx[2][4 + i] += v * w2;
        }
    }

    // Gaunt contraction per channel, scatter tp into extra slots
#pragma unroll
    for (int ch = 0; ch < 3; ++ch) {
        float wy0 = Wys[2 * ch], wy1 = Wys[2 * ch + 1];
        float cy[9];
        cy[0] = 0.0f;
#pragma unroll
        for (int k = 1; k < 4; ++k) cy[k] = wy0 * sh[k];
#pragma unroll
        for (int k = 4; k < 9; ++k) cy[k] = wy1 * sh[k];
        float tp[9];
#pragma unroll
        for (int k = 0; k < 9; ++k) tp[k] = 0.0f;
        for (int k1 = 0; k1 < 9; ++k1) {
            float a = cx[ch][k1];
            for (int k2 = 1; k2 < 9; ++k2) {
                float p = a * cy[k2];
                const float* g = &Gs[(k1 * 9 + k2) * 9];
#pragma unroll
                for (int k3 = 0; k3 < 9; ++k3) tp[k3] += p * g[k3];
            }
        }
        atomicAdd(&ag[64 + ch], tp[0] * inv12);
#pragma unroll
        for (int i = 0; i < 3; ++i) atomicAdd(&ag[67 + 3 * (32 + ch) + i], tp[1 + i] * inv12);
#pragma unroll
        for (int i = 0; i < 5; ++i) atomicAdd(&ag[172 + 5 * (16 + ch) + i], tp[4 + i] * inv12);
    }
}

// ---------------------------------------------------------------------------
// WMMA f32 16x16x4 GEMM tile helper.
// D[m][n] = sum_k A[m][k]*B[k][n], one wave computes a 16x16 tile.
// A frag: lanes 0-15 hold rows M=0..15 at K={k0,k0+1}; lanes 16-31 at K={k0+2,k0+3}.
// B frag: lanes 0-15 hold cols N=0..15 at same K split.
// D: vgpr v, lane L -> (m = v + 8*(L>>4), n = L&15).
// ---------------------------------------------------------------------------
template <class LA, class LB>
__device__ __forceinline__ v8f gemm_tile(int K, LA loadA, LB loadB) {
    int lane = threadIdx.x & 31;
    int half = lane >> 4;
    int mr = lane & 15;
    v8f acc;
#pragma unroll
    for (int v = 0; v < 8; ++v) acc[v] = 0.0f;
#ifdef USE_WMMA_F32
    for (int k0 = 0; k0 < K; k0 += 4) {
        int ka = k0 + 2 * half;
        v2f A, B;
        A.x = (ka < K) ? loadA(mr, ka) : 0.0f;
        A.y = (ka + 1 < K) ? loadA(mr, ka + 1) : 0.0f;
        B.x = (ka < K) ? loadB(ka, mr) : 0.0f;
        B.y = (ka + 1 < K) ? loadB(ka + 1, mr) : 0.0f;
        acc = __builtin_amdgcn_wmma_f32_16x16x4_f32(false, A, false, B, (short)0, acc,
                                                    false, false);
    }
#else
    for (int k = 0; k < K; ++k) {
        float b = loadB(k, mr);
#pragma unroll
        for (int v = 0; v < 8; ++v) acc[v] += loadA(v + 8 * half, k) * b;
    }
#endif
    return acc;
}

__device__ __forceinline__ float gelu_tanh(float v) {
    const float k0 = 0.7978845608028654f;  // sqrt(2/pi)
    float inner = k0 * (v + 0.044715f * v * v * v);
    return 0.5f * v * (1.0f + tanhf(inner));
}

// ---------------------------------------------------------------------------
// Kernel 3: per-16-node-tile dense layers, one wave per tile, all via WMMA.
// The 1/12 message normalization is folded into the post-GEMM scale so the
// A-fragment loads feed v_wmma directly (no per-K v_pk_mul on the chain).
// ---------------------------------------------------------------------------
__global__ void node_kernel(const float* __restrict__ x, const float* __restrict__ agg,
                            const float* __restrict__ Wp0, const float* __restrict__ Wp1,
                            const float* __restrict__ Wp2, const float* __restrict__ Wq0,
                            const float* __restrict__ Wq1, const float* __restrict__ Wq2,
                            const float* __restrict__ Ws0, const float* __restrict__ Ws1,
                            const float* __restrict__ Ws2, float* __restrict__ out, int N) {
    __shared__ float h[16 * 240];  // [m][0:64]=h0, [m][64+o*3+i]=h1, [m][160+o*5+i]=h2
    const int n0 = blockIdx.x * 16;
    const int lane = threadIdx.x & 31;
    const int half = lane >> 4;
    const int mr = lane & 15;
    // stage-1 scales include the 1/12 aggregation normalization
    const float s67 = 0.12216944435630522f / 12.0f;  // 1/(12*sqrt(67))
    const float s35 = 0.16903085094570331f / 12.0f;  // 1/(12*sqrt(35))
    const float s19 = 0.22941573387056174f / 12.0f;  // 1/(12*sqrt(19))
    const float is64 = 0.125f;
    const float is32 = 0.17677669529663687f;  // 1/sqrt(32)
    const float is16 = 0.25f;

    auto nodeOf = [&](int m) { int n = n0 + m; return (n < N) ? n : (N - 1); };

    // ---- stage 1: h = Wp(agg/12)/sqrt(c) (gelu on l=0) ----
    for (int ot = 0; ot < 4; ++ot) {  // h0: K=67 -> 64 outs
        v8f acc = gemm_tile(67,
            [&](int m, int k) { return agg[(size_t)nodeOf(m) * 267 + k]; },
            [&](int k, int n) { return Wp0[k * 64 + ot * 16 + n]; });
#pragma unroll
        for (int v = 0; v < 8; ++v)
            h[(v + 8 * half) * 240 + ot * 16 + mr] = gelu_tanh(acc[v] * s67);
    }
#pragma unroll
    for (int i = 0; i < 3; ++i)  // h1: K=35 -> 32 outs, per i
        for (int ot = 0; ot < 2; ++ot) {
            v8f acc = gemm_tile(35,
                [&](int m, int k) { return agg[(size_t)nodeOf(m) * 267 + 67 + k * 3 + i]; },
                [&](int k, int n) { return Wp1[k * 32 + ot * 16 + n]; });
#pragma unroll
            for (int v = 0; v < 8; ++v)
                h[(v + 8 * half) * 240 + 64 + (ot * 16 + mr) * 3 + i] = acc[v] * s35;
        }
#pragma unroll
    for (int i = 0; i < 5; ++i) {  // h2: K=19 -> 16 outs, per i
        v8f acc = gemm_tile(19,
            [&](int m, int k) { return agg[(size_t)nodeOf(m) * 267 + 172 + k * 5 + i]; },
            [&](int k, int n) { return Wp2[k * 16 + n]; });
#pragma unroll
        for (int v = 0; v < 8; ++v)
            h[(v + 8 * half) * 240 + 160 + mr * 5 + i] = acc[v] * s19;
    }

    // ---- stage 2: out = Wq(h)/sqrt(c) + Ws(x)/sqrt(c) ----
    for (int ot = 0; ot < 4; ++ot) {  // l=0
        v8f g = gemm_tile(64,
            [&](int m, int k) { return h[m * 240 + k]; },
            [&](int k, int n) { return Wq0[k * 64 + ot * 16 + n]; });
        v8f s = gemm_tile(64,
            [&](int m, int k) { return x[(size_t)nodeOf(m) * 240 + k]; },
            [&](int k, int n) { return Ws0[k * 64 + ot * 16 + n]; });
#pragma unroll
        for (int v = 0; v < 8; ++v) {
            int m = v + 8 * half;
            if (n0 + m < N)
                out[(size_t)(n0 + m) * 240 + ot * 16 + mr] = (g[v] + s[v]) * is64;
        }
    }
#pragma unroll
    for (int i = 0; i < 3; ++i)  // l=1
        for (int ot = 0; ot < 2; ++ot) {
            v8f g = gemm_tile(32,
                [&](int m, int k) { return h[m * 240 + 64 + k * 3 + i]; },
                [&](int k, int n) { return Wq1[k * 32 + ot * 16 + n]; });
            v8f s = gemm_tile(32,
                [&](int m, int k) { return x[(size_t)nodeOf(m) * 240 + 64 + k * 3 + i]; },
                [&](int k, int n) { return Ws1[k * 32 + ot * 16 + n]; });
#pragma unroll
            for (int v = 0; v < 8; ++v) {
                int m = v + 8 * half;
                if (n0 + m < N)
                    out[(size_t)(n0 + m) * 240 + 64 + (ot * 16 + mr) * 3 + i] =
                        (g[v] + s[v]) * is32;
            }
        }
#pragma unroll
    for (int i = 0; i < 5; ++i) {  // l=2
        v8f g = gemm_tile(16,
            [&](int m, int k) { return h[m * 240 + 160 + k * 5 + i]; },
            [&](int k, int n) { return Wq2[k * 16 + n]; });
        v8f s = gemm_tile(16,
            [&](int m, int k) { return x[(size_t)nodeOf(m) * 240 + 160 + k * 5 + i]; },
            [&](int k, int n) { return Ws2[k * 16 + n]; });
#pragma unroll
        for (int v = 0; v < 8; ++v) {
            int m = v + 8 * half;
            if (n0 + m < N)
                out[(size_t)(n0 + m) * 240 + 160 + mr * 5 + i] = (g[v] + s[v]) * is16;
        }
    }
}

// ---------------------------------------------------------------------------
extern "C" void kernel_launch(void* const* d_in, const int* in_sizes, int n_in,
                              void* d_out, int out_size, void* d_ws, size_t ws_size,
                              hipStream_t stream) {
    const float* x    = (const float*)d_in[0];
    const float* pos  = (const float*)d_in[1];
    const float* Wx   = (const float*)d_in[2];
    const float* Wy   = (const float*)d_in[3];
    const float* Wp0  = (const float*)d_in[4];
    const float* Wp1  = (const float*)d_in[5];
    const float* Wp2  = (const float*)d_in[6];
    const float* Wq0  = (const float*)d_in[7];
    const float* Wq1  = (const float*)d_in[8];
    const float* Wq2  = (const float*)d_in[9];
    const float* Ws0  = (const float*)d_in[10];
    const float* Ws1  = (const float*)d_in[11];
    const float* Ws2  = (const float*)d_in[12];
    const int* senders   = (const int*)d_in[13];
    const int* receivers = (const int*)d_in[14];

    const int N = in_sizes[0] / 240;
    const int E = in_sizes[13];

    float* G   = (float*)d_ws;                       // 729 floats
    float* agg = (float*)((char*)d_ws + 4096);       // N*267 floats

    hipMemsetAsync(agg, 0, (size_t)N * 267 * sizeof(float), stream);
    init_basis_kernel<<<1, 256, 0, stream>>>(G);
    edge_kernel<<<(E + 255) / 256, 256, 0, stream>>>(x, pos, Wx, Wy, senders, receivers,
                                                     G, agg, E);
    node_kernel<<<(N + 15) / 16, 32, 0, stream>>>(x, agg, Wp0, Wp1, Wp2, Wq0, Wq1, Wq2,
                                                  Ws0, Ws1, Ws2, (float*)d_out, N);
}